// SpERT_4148938408933
// MI455X (gfx1250) — compile-verified
//
#include <hip/hip_runtime.h>
#include <hip/hip_bf16.h>

// SpERT head for MI455X (gfx1250, wave32).
// Phase 1: per-span masked max-pool (+ ctx copy + size-emb gather) -> entity_repr in ws
// Phase 2: per-relation masked max-pool + pair gather -> rel_repr in ws
// Phase 3: skinny f32 GEMMs via V_WMMA_F32_16X16X4_F32 (exact fp32 path)

#define Bsz 4
#define Lsz 512
#define Hsz 768
#define Ssz 100
#define Rsz 100
#define Esz 128
#define TEsz 8
#define TRsz 6
#define KE (2 * Hsz + Esz)      // 1664 (entity_repr width)  == span_w rows
#define KR (3 * Hsz + 2 * Esz)  // 2560 (relation_repr width) == rel_w rows
#define NEGF (-1e20f)

typedef __attribute__((ext_vector_type(2))) float v2f;
typedef __attribute__((ext_vector_type(8))) float v8f;

// ---------------------------------------------------------------------------
// Kernel 1: entity masked-max pool. One block per (b,s) span, 256 threads.
// Writes entity_repr[row] = [pooled(768) | ctx(768) | size_emb[len](128)]
// ---------------------------------------------------------------------------
__global__ __launch_bounds__(256) void spert_entity_pool(
    const float* __restrict__ lhs,      // (B,L,H)
    const int* __restrict__ emask,      // (B,S,L)
    const float* __restrict__ size_emb, // (100,E)
    float* __restrict__ erepr)          // (B*S, KE)
{
  const int row = blockIdx.x;           // b*S + s
  const int b = row / Ssz;
  const int t = threadIdx.x;

  __shared__ int smask[Lsz];
  __shared__ int partial[256];

  const int* m = emask + (size_t)row * Lsz;
  smask[t] = m[t];
  smask[t + 256] = m[t + 256];
  __syncthreads();

  // span length = sum(mask)
  partial[t] = smask[t] + smask[t + 256];
  __syncthreads();
  for (int off = 128; off > 0; off >>= 1) {
    if (t < off) partial[t] += partial[t + off];
    __syncthreads();
  }
  const int len = partial[0];

  // masked max over L, each thread owns dims {t, t+256, t+512}
  const float* hb = lhs + (size_t)b * Lsz * Hsz;
  float m0 = NEGF, m1 = NEGF, m2 = NEGF;
  for (int l = 0; l < Lsz; ++l) {
    if (smask[l]) {  // uniform branch across block
      const float* hr = hb + (size_t)l * Hsz;
      m0 = fmaxf(m0, hr[t]);
      m1 = fmaxf(m1, hr[t + 256]);
      m2 = fmaxf(m2, hr[t + 512]);
    }
  }

  float* out = erepr + (size_t)row * KE;
  out[t] = m0;
  out[t + 256] = m1;
  out[t + 512] = m2;
  // context = lhs[b, 0, :]
  out[Hsz + t] = hb[t];
  out[Hsz + t + 256] = hb[t + 256];
  out[Hsz + t + 512] = hb[t + 512];
  // size embedding
  if (t < Esz) out[2 * Hsz + t] = size_emb[(size_t)len * Esz + t];
}

// ---------------------------------------------------------------------------
// Kernel 2: relation context pool + entity-pair gather. One block per (b,r).
// rel_repr[row] = [rel_ctx(768) | head(768) | tail(768) | hsize(128) | tsize(128)]
// ---------------------------------------------------------------------------
__global__ __launch_bounds__(256) void spert_rel_pool(
    const float* __restrict__ lhs,        // (B,L,H)
    const int* __restrict__ rmask,        // (B,R,L)
    const int* __restrict__ rsample,      // (B,R)
    const int* __restrict__ relations,    // (B,R,2)
    const float* __restrict__ erepr,      // (B*S, KE)
    float* __restrict__ rrepr)            // (B*R, KR)
{
  const int row = blockIdx.x;             // b*R + r
  const int b = row / Rsz;
  const int t = threadIdx.x;

  __shared__ int smask[Lsz];
  const int* m = rmask + (size_t)row * Lsz;
  smask[t] = m[t];
  smask[t + 256] = m[t + 256];
  __syncthreads();

  const float* hb = lhs + (size_t)b * Lsz * Hsz;
  float m0 = NEGF, m1 = NEGF, m2 = NEGF;
  for (int l = 0; l < Lsz; ++l) {
    if (smask[l]) {
      const float* hr = hb + (size_t)l * Hsz;
      m0 = fmaxf(m0, hr[t]);
      m1 = fmaxf(m1, hr[t + 256]);
      m2 = fmaxf(m2, hr[t + 512]);
    }
  }
  if (rsample[row] == 1) { m0 = 0.f; m1 = 0.f; m2 = 0.f; }

  float* out = rrepr + (size_t)row * KR;
  out[t] = m0;
  out[t + 256] = m1;
  out[t + 512] = m2;

  const int head = relations[(size_t)row * 2 + 0];
  const int tail = relations[(size_t)row * 2 + 1];
  const float* eh = erepr + (size_t)(b * Ssz + head) * KE;
  const float* et = erepr + (size_t)(b * Ssz + tail) * KE;

  out[Hsz + t] = eh[t];
  out[Hsz + t + 256] = eh[t + 256];
  out[Hsz + t + 512] = eh[t + 512];
  out[2 * Hsz + t] = et[t];
  out[2 * Hsz + t + 256] = et[t + 256];
  out[2 * Hsz + t + 512] = et[t + 512];
  if (t < Esz) {
    out[3 * Hsz + t] = eh[2 * Hsz + t];          // head size emb
    out[3 * Hsz + Esz + t] = et[2 * Hsz + t];    // tail size emb
  }
}

// ---------------------------------------------------------------------------
// Kernel 3: skinny GEMM  out(M,ncols) = A(M,K) @ W(K,ncols) + bias
// via V_WMMA_F32_16X16X4_F32. One wave (32 lanes) per 16-row tile; N padded
// to 16 with zero columns. M must be a multiple of 16 (400 = 25*16).
//
// ISA layouts (CDNA5 ISA):
//   A 16x4:  lanes 0-15 -> M=lane, VGPR{0,1}=K{0,1}; lanes 16-31 -> K{2,3}
//   B 4x16:  VGPR0: lanes0-15 row K0, lanes16-31 row K2; VGPR1: K1 / K3
//   D 16x16: VGPR v: lanes0-15 M=v, lanes16-31 M=v+8; N = lane&15
//
// B loads are branch-free: lanes with N >= ncols read a clamped (valid)
// address and multiply by 0, so the K-loop has no EXEC-mask toggling and
// EXEC stays all-1s through every v_wmma (ISA requirement).
// ---------------------------------------------------------------------------
__global__ __launch_bounds__(32) void spert_wmma_gemm(
    const float* __restrict__ A,
    const float* __restrict__ W,     // (K, ncols) row-major
    const float* __restrict__ bias,  // (ncols)
    float* __restrict__ out,         // (M, ncols) row-major
    int K, int ncols)
{
  const int tile = blockIdx.x;
  const int lane = threadIdx.x;
  const int mlo = lane & 15;
  const int khi = lane >> 4;                  // 0 = lanes 0-15, 1 = lanes 16-31
  const float* arow = A + (size_t)(tile * 16 + mlo) * K + 2 * khi;
  const bool ncol_ok = (mlo < ncols);
  const int ncol = ncol_ok ? mlo : 0;         // clamped, always in-bounds
  const float bscale = ncol_ok ? 1.f : 0.f;   // zero out padded N columns
  const float* wcol = W + (size_t)2 * khi * ncols + ncol;

  v8f c = {0.f, 0.f, 0.f, 0.f, 0.f, 0.f, 0.f, 0.f};

  for (int k = 0; k < K; k += 4) {
    v2f av, bv;
    av.x = arow[k + 0];                                   // K = k + 2*khi
    av.y = arow[k + 1];                                   // K = k + 2*khi + 1
    bv.x = wcol[(size_t)k * ncols] * bscale;              // row k + 2*khi
    bv.y = wcol[(size_t)k * ncols + ncols] * bscale;      // row k + 2*khi + 1
    // 8 args: (neg_a, A, neg_b, B, c_mod, C, reuse_a, reuse_b)
    c = __builtin_amdgcn_wmma_f32_16x16x4_f32(
        false, av, false, bv, (short)0, c, false, false);
  }

  if (ncol_ok) {
    const float bb = bias[mlo];
    for (int v = 0; v < 8; ++v) {
      const int rowm = tile * 16 + v + 8 * khi;
      out[(size_t)rowm * ncols + mlo] = c[v] + bb;
    }
  }
}

// ---------------------------------------------------------------------------
extern "C" void kernel_launch(void* const* d_in, const int* in_sizes, int n_in,
                              void* d_out, int out_size, void* d_ws, size_t ws_size,
                              hipStream_t stream) {
  (void)in_sizes; (void)n_in; (void)out_size; (void)ws_size;

  const float* lhs       = (const float*)d_in[0];  // (B,L,H)
  const int*   emask     = (const int*)d_in[1];    // (B,S,L)
  const int*   relations = (const int*)d_in[2];    // (B,R,2)
  const int*   rmask     = (const int*)d_in[3];    // (B,R,L)
  const int*   rsample   = (const int*)d_in[4];    // (B,R)
  const float* size_emb  = (const float*)d_in[5];  // (100,E)
  const float* span_w    = (const float*)d_in[6];  // (KE,TE)
  const float* span_b    = (const float*)d_in[7];  // (TE)
  const float* rel_w     = (const float*)d_in[8];  // (KR,TR)
  const float* rel_b     = (const float*)d_in[9];  // (TR)

  float* out = (float*)d_out;  // entity_logit (B*S*TE) then relation_logit (B*R*TR)

  // workspace: entity_repr (400*1664 f32 = 2.66MB) then rel_repr (400*2560 f32 = 4.1MB)
  float* erepr = (float*)d_ws;
  float* rrepr = erepr + (size_t)Bsz * Ssz * KE;

  spert_entity_pool<<<Bsz * Ssz, 256, 0, stream>>>(lhs, emask, size_emb, erepr);
  spert_rel_pool<<<Bsz * Rsz, 256, 0, stream>>>(lhs, rmask, rsample, relations,
                                                erepr, rrepr);

  // entity logits: M=400 (25 tiles), K=1664, N=8
  spert_wmma_gemm<<<(Bsz * Ssz) / 16, 32, 0, stream>>>(erepr, span_w, span_b,
                                                       out, KE, TEsz);
  // relation logits: M=400 (25 tiles), K=2560, N=6
  spert_wmma_gemm<<<(Bsz * Rsz) / 16, 32, 0, stream>>>(
      rrepr, rel_w, rel_b, out + (size_t)Bsz * Ssz * TEsz, KR, TRsz);
}